// RPN_12103217840575
// MI455X (gfx1250) — compile-verified
//
#include <hip/hip_runtime.h>
#include <hip/hip_bf16.h>

// Problem constants (match reference setup_inputs()).
#define B_    8
#define A_    184320
#define G_    64
#define BLK   256
#define ITEMS 4          // anchors per thread (blocked -> lane order == index order)
#define FG_T  0.7f
#define BG_T  0.3f
#define IMG_  640.0f

typedef unsigned int       u32;
typedef unsigned long long u64;
typedef unsigned int v4u __attribute__((ext_vector_type(4)));
typedef int          v4i __attribute__((ext_vector_type(4)));
typedef int          v8i __attribute__((ext_vector_type(8)));

// 5-step wave32 butterfly max on a packed (iou_bits<<32 | ~anchor_idx) key.
// iou >= 0 always, so IEEE bit pattern is monotonic; ~idx => ties pick lowest index.
__device__ __forceinline__ u64 wave32_max_u64(u64 key) {
#pragma unroll
  for (int off = 16; off > 0; off >>= 1) {
    u64 o = __shfl_xor(key, off, 32);
    key = (o > key) ? o : key;
  }
  return key;
}

// d_out layout (floats): labels [B*A] | matched [B*A*4] | max_iou [B*A]
// (int32 labels promote to f32 when the reference tuple is concatenated flat).

__global__ __launch_bounds__(BLK) void rpn_assign_12103217840575(
    const float* __restrict__ anchors,   // [B,A,4] cxcywh
    const float* __restrict__ gt,        // [B,G,4] xyxy
    u64*         __restrict__ gwin,      // [B,G] packed per-gt best-anchor keys
    float*       __restrict__ out)
{
  __shared__ float raw[G_ * 4];                       // TDM destination (AoS gt tile)
  __shared__ float gx1[G_], gy1[G_], gx2[G_], gy2[G_], gar[G_];
  __shared__ u64   win[G_];                           // block-local per-gt best keys

  const int b = blockIdx.y;

  // ---- TDM: DMA this batch's 64x4 f32 gt tile into LDS (wave 0 only) ----
  if (threadIdx.x < 32) {
    u64 ga  = (u64)(uintptr_t)(gt + (size_t)b * G_ * 4);
    u32 lds = (u32)(uintptr_t)(void*)&raw[0];         // low 32 bits = LDS byte offset
    // D# group0: count=1 | lds_addr | global_addr[56:0] | type=2 (bits 127:126)
    v4u g0 = { 0x1u, lds, (u32)ga,
               (((u32)(ga >> 32)) & 0x01FFFFFFu) | 0x80000000u };
    // D# group1: data_size=4B; tensor_dim0=256, tensor_dim1=1; tile_dim0=256,
    // tile_dim1=1, tile_dim2=0; tensor_dim0_stride=256; tensor_dim1_stride=256.
    v8i g1 = { (int)0x00020000u,   // workgroup_mask=0, data_size=2 (4B)
               (int)0x01000000u,   // tensor_dim0[15:0]=256 in [31:16]
               (int)0x00010000u,   // tensor_dim1[15:0]=1  in [31:16]
               (int)0x01000000u,   // tile_dim0=256        in [31:16]
               1,                  // tile_dim1=1, tile_dim2=0
               256,                // tensor_dim0_stride[31:0]
               (int)0x01000000u,   // tensor_dim1_stride[15:0]=256 in [31:16]
               0 };
    v4i z4 = { 0, 0, 0, 0 };
    v8i z8 = { 0, 0, 0, 0, 0, 0, 0, 0 };
    // 6-arg toolchain variant: (g0, g1, g2, g3, g4, cpol)
    __builtin_amdgcn_tensor_load_to_lds(g0, g1, z4, z4, z8, 0);
    __builtin_amdgcn_s_wait_tensorcnt(0);
  }
  __syncthreads();

  // Unpack gt tile AoS->SoA + precompute areas; init block winner table.
  if (threadIdx.x < G_) {
    const int g = threadIdx.x;
    const float x1 = raw[g * 4 + 0], y1 = raw[g * 4 + 1];
    const float x2 = raw[g * 4 + 2], y2 = raw[g * 4 + 3];
    gx1[g] = x1; gy1[g] = y1; gx2[g] = x2; gy2[g] = y2;
    gar[g] = (x2 - x1) * (y2 - y1);
    win[g] = 0ull;
  }
  __syncthreads();

  // ---- load 4 anchors/thread, convert cxcywh->xyxy, precompute area & cross ----
  const int base = blockIdx.x * (BLK * ITEMS) + threadIdx.x * ITEMS;  // A_ divides evenly
  float ax1[ITEMS], ay1[ITEMS], ax2[ITEMS], ay2[ITEMS], aar[ITEMS];
  float best[ITEMS];
  int   bestg[ITEMS];
  bool  crossf[ITEMS];

  const float4* ap = (const float4*)anchors + (size_t)b * A_ + base;
#pragma unroll
  for (int i = 0; i < ITEMS; ++i) {
    const float4 c = ap[i];
    const float hw = 0.5f * c.z, hh = 0.5f * c.w;
    ax1[i] = c.x - hw; ay1[i] = c.y - hh;
    ax2[i] = c.x + hw; ay2[i] = c.y + hh;
    aar[i]  = c.z * c.w;
    best[i] = 0.0f;     // iou >= 0; first-tie => g stays 0 (matches jnp.argmax)
    bestg[i] = 0;
    crossf[i] = (ax1[i] < 0.0f) | (ay1[i] < 0.0f) | (ax2[i] > IMG_) | (ay2[i] > IMG_);
  }

  // ---- main IoU loop over 64 gts ----
  for (int g = 0; g < G_; ++g) {
    const float x1 = gx1[g], y1 = gy1[g], x2 = gx2[g], y2 = gy2[g], ag = gar[g];
    float tmax = 0.0f;
    u32   tidx = (u32)base;   // first-tie at 0.0 => lowest index of this thread
#pragma unroll
    for (int i = 0; i < ITEMS; ++i) {
      const float iw    = fminf(ax2[i], x2) - fmaxf(ax1[i], x1);
      const float ih    = fminf(ay2[i], y2) - fmaxf(ay1[i], y1);
      const float inter = fmaxf(iw, 0.0f) * fmaxf(ih, 0.0f);
      const float den   = aar[i] + ag - inter;
      const float iou   = inter * __builtin_amdgcn_rcpf(den);
      if (iou > best[i]) { best[i] = iou; bestg[i] = g; }   // first-index ties
      if (iou > tmax)    { tmax = iou; tidx = (u32)(base + i); }
    }
    // per-gt argmax over anchors: wave reduce, then one LDS u64 max per wave
    u64 key = ((u64)__float_as_uint(tmax) << 32) | (u32)(~tidx);
    key = wave32_max_u64(key);
    if ((threadIdx.x & 31) == 0) atomicMax(&win[g], key);
  }

  // ---- per-anchor outputs (rules 2, 3 and cross filter; rule 1 fixed up later) ----
  float* labels  = out;
  float* matched = out + (size_t)B_ * A_;
  float* maxiou  = out + (size_t)5 * B_ * A_;
  const size_t row = (size_t)b * A_ + base;
#pragma unroll
  for (int i = 0; i < ITEMS; ++i) {
    const float mv = best[i];
    const bool  fg = mv > FG_T;
    float lab = fg ? 1.0f : ((mv < BG_T) ? 0.0f : -1.0f);
    if (crossf[i]) lab = -1.0f;
    labels[row + i] = lab;
    maxiou[row + i] = mv;
    float4 m = { 0.0f, 0.0f, 0.0f, 0.0f };
    if (fg) {
      const int g = bestg[i];
      m = make_float4(gx1[g], gy1[g], gx2[g], gy2[g]);
    }
    ((float4*)matched)[row + i] = m;
  }

  // ---- merge block winners into the global per-(b,g) table ----
  __syncthreads();
  if (threadIdx.x < G_) atomicMax(&gwin[(size_t)b * G_ + threadIdx.x], win[threadIdx.x]);
}

// Rule 1 fixup: each gt's best anchor gets label 1 (unless cross) and, if that
// anchor is not already fg (rule 2 overrides matched), matched = this gt box.
__global__ void rpn_finalize_12103217840575(
    const float* __restrict__ anchors,
    const float* __restrict__ gt,
    const u64*   __restrict__ gwin,
    float*       __restrict__ out)
{
  const int b = blockIdx.x;
  const int g = threadIdx.x;
  const u64 key = gwin[(size_t)b * G_ + g];
  const u32 a = ~((u32)key);
  if (key == 0ull || a >= A_) return;

  const float* apx = anchors + ((size_t)b * A_ + a) * 4;
  const float cx = apx[0], cy = apx[1], w = apx[2], h = apx[3];
  const float x1 = cx - 0.5f * w, y1 = cy - 0.5f * h;
  const float x2 = cx + 0.5f * w, y2 = cy + 0.5f * h;
  const bool cross = (x1 < 0.0f) | (y1 < 0.0f) | (x2 > IMG_) | (y2 > IMG_);

  float* labels  = out;
  float* matched = out + (size_t)B_ * A_;
  float* maxiou  = out + (size_t)5 * B_ * A_;

  const float mv = maxiou[(size_t)b * A_ + a];
  const bool  fg = mv > FG_T;
  if (!cross) labels[(size_t)b * A_ + a] = 1.0f;
  if (!fg) {
    const float* gp = gt + ((size_t)b * G_ + g) * 4;
    ((float4*)matched)[(size_t)b * A_ + a] = make_float4(gp[0], gp[1], gp[2], gp[3]);
  }
}

extern "C" void kernel_launch(void* const* d_in, const int* in_sizes, int n_in,
                              void* d_out, int out_size, void* d_ws, size_t ws_size,
                              hipStream_t stream) {
  const float* anchors = (const float*)d_in[0];   // [B,A,4] f32
  const float* gt      = (const float*)d_in[1];   // [B,G,4] f32
  float*       out     = (float*)d_out;
  u64*         gwin    = (u64*)d_ws;              // [B*G] packed winner keys

  (void)in_sizes; (void)n_in; (void)out_size; (void)ws_size;

  (void)hipMemsetAsync(d_ws, 0, (size_t)B_ * G_ * sizeof(u64), stream);

  dim3 grid(A_ / (BLK * ITEMS), B_);
  rpn_assign_12103217840575<<<grid, dim3(BLK), 0, stream>>>(anchors, gt, gwin, out);
  rpn_finalize_12103217840575<<<dim3(B_), dim3(G_), 0, stream>>>(anchors, gt, gwin, out);
}